// GNN_10883447128785
// MI455X (gfx1250) — compile-verified
//
#include <hip/hip_runtime.h>
#include <hip/hip_bf16.h>
#include <math.h>

typedef __attribute__((ext_vector_type(2))) float v2f;
typedef __attribute__((ext_vector_type(8))) float v8f;

// ---------------------------------------------------------------------------
// WMMA f32 16x16x4 wrapper (full fp32 precision, matches reference numerics).
// A: 16x4, lane(half=lane>>4, r=lane&15): v0=A[r][2h], v1=A[r][2h+1]
// B: 4x16:                                 v0=B[2h][r], v1=B[2h+1][r]
// C/D: 8 VGPRs, element v = C[v + 8h][r]
// ---------------------------------------------------------------------------
__device__ __forceinline__ v8f wmma4(v2f a, v2f b, v8f c) {
  return __builtin_amdgcn_wmma_f32_16x16x4_f32(false, a, false, b, (short)0, c,
                                               false, false);
}

// C[n x 64] = act( [A0 | A1] (n x (K0+K1)) @ W^T + bias ), W row-major [64 x K]
// act: 0 = relu, 1 = sigmoid. Optionally mirrors the result into C2.
// One wave computes a full 16-row x 64-col strip (4 accumulators, A reuse x4).
__global__ __launch_bounds__(256) void k_gemm64(
    const float* __restrict__ A0, int K0, const float* __restrict__ A1, int K1,
    const float* __restrict__ W, const float* __restrict__ bias,
    float* __restrict__ C, float* __restrict__ C2, int n, int act) {
  int gw = (blockIdx.x * blockDim.x + threadIdx.x) >> 5;  // wave id = row tile
  int lane = threadIdx.x & 31;
  if (gw * 16 >= n) return;
  const int half = lane >> 4, r = lane & 15;
  const int K = K0 + K1;
  const int row = gw * 16 + r;
  v8f acc0 = {}, acc1 = {}, acc2 = {}, acc3 = {};
  for (int kb = 0; kb < K; kb += 4) {
    int kl = kb + 2 * half;
    float a0, a1;
    if (kl < K0) {  // K0 is a multiple of 4, so (kl,kl+1) never straddles
      a0 = A0[(size_t)row * K0 + kl];
      a1 = A0[(size_t)row * K0 + kl + 1];
    } else {
      a0 = A1[(size_t)row * K1 + (kl - K0)];
      a1 = A1[(size_t)row * K1 + (kl - K0) + 1];
    }
    v2f av; av.x = a0; av.y = a1;
    const float* wp = W + (size_t)r * K + kl;  // + nt*16*K per col tile
    v2f b0; b0.x = wp[0];          b0.y = wp[1];
    v2f b1; b1.x = wp[16 * K];     b1.y = wp[16 * K + 1];
    v2f b2; b2.x = wp[32 * K];     b2.y = wp[32 * K + 1];
    v2f b3; b3.x = wp[48 * K];     b3.y = wp[48 * K + 1];
    acc0 = wmma4(av, b0, acc0);
    acc1 = wmma4(av, b1, acc1);
    acc2 = wmma4(av, b2, acc2);
    acc3 = wmma4(av, b3, acc3);
  }
  v8f accs[4] = {acc0, acc1, acc2, acc3};
#pragma unroll
  for (int nt = 0; nt < 4; ++nt) {
    int col = nt * 16 + r;
    float bv = bias ? bias[col] : 0.f;
#pragma unroll
    for (int v = 0; v < 8; ++v) {
      float val = accs[nt][v] + bv;
      val = (act == 0) ? fmaxf(val, 0.f) : 1.f / (1.f + expf(-val));
      size_t off = (size_t)(gw * 16 + v + 8 * half) * 64 + col;
      C[off] = val;
      if (C2) C2[off] = val;
    }
  }
}

// x[i,d] = emb[ids[i], d]
__global__ void k_embed(const int* __restrict__ ids,
                        const float* __restrict__ emb, float* __restrict__ x,
                        int total) {
  int t = blockIdx.x * blockDim.x + threadIdx.x;
  if (t >= total) return;
  x[t] = emb[ids[t >> 6] * 64 + (t & 63)];
}

__global__ void k_init_edges(const int* __restrict__ s, const int* __restrict__ d,
                             int* __restrict__ so, int* __restrict__ dso,
                             int* __restrict__ em, int E) {
  int t = blockIdx.x * blockDim.x + threadIdx.x;
  if (t >= E) return;
  so[t] = s[t]; dso[t] = d[t]; em[t] = 1;
}

// Scatter max over edges. All msg values >= 0 (post-relu) and aggr is
// initialized to msg, so unsigned-int ordering == float ordering: use the
// native deterministic GLOBAL_ATOMIC_MAX_U32 path.
__global__ void k_edge_max(const int* __restrict__ src,
                           const int* __restrict__ dst,
                           const int* __restrict__ em,
                           const float* __restrict__ msg,
                           float* __restrict__ aggr, long long total) {
  long long t = blockIdx.x * (long long)blockDim.x + threadIdx.x;
  if (t >= total) return;
  int e = (int)(t >> 4);
  if (!em[e]) return;
  int q = ((int)t & 15) * 4;
  const float* ms = msg + (size_t)src[e] * 64 + q;
  unsigned* ag = (unsigned*)aggr + (size_t)dst[e] * 64 + q;
#pragma unroll
  for (int j = 0; j < 4; ++j)
    atomicMax(&ag[j], __float_as_uint(ms[j]));
}

// score[i] = dot(x[i,:], p) / ||p||, one wave32 per node
__global__ void k_score(const float* __restrict__ x, const float* __restrict__ p,
                        float* __restrict__ sc, int n) {
  int wid = (blockIdx.x * blockDim.x + threadIdx.x) >> 5;
  int lane = threadIdx.x & 31;
  if (wid >= n) return;
  float p0 = p[lane], p1 = p[lane + 32];
  float v = x[(size_t)wid * 64 + lane] * p0 + x[(size_t)wid * 64 + 32 + lane] * p1;
  float q = p0 * p0 + p1 * p1;
  for (int o = 16; o > 0; o >>= 1) {
    v += __shfl_xor(v, o, 32);
    q += __shfl_xor(q, o, 32);
  }
  if (lane == 0) sc[wid] = v / sqrtf(q);
}

// One block per graph: bitonic sort 1024 (score,idx) pairs in LDS
// (desc score, asc idx on ties == jax.lax.top_k set), mark keepers,
// prefix-sum compaction in original index order (output order-invariant).
__global__ __launch_bounds__(512) void k_topk(const float* __restrict__ sc,
                                              int nPer, int kOut,
                                              int* __restrict__ inv,
                                              int* __restrict__ perm) {
  __shared__ float s[1024];
  __shared__ int si[1024];
  __shared__ int fA[1024];
  __shared__ int fB[1024];
  int g = blockIdx.x, tid = threadIdx.x;
  for (int i = tid; i < 1024; i += 512) {
    if (i < nPer) { s[i] = sc[g * nPer + i]; si[i] = i; }
    else          { s[i] = -3.402823466e38f; si[i] = 0x7fffffff; }
  }
  for (int kk = 2; kk <= 1024; kk <<= 1) {
    for (int j = kk >> 1; j > 0; j >>= 1) {
      __syncthreads();
      int ix = ((tid & ~(j - 1)) << 1) | (tid & (j - 1));
      int l = ix | j;
      float sa = s[ix], sb = s[l];
      int ia = si[ix], ib = si[l];
      bool aFirst = (sa > sb) || (sa == sb && ia < ib);
      bool dirDesc = ((ix & kk) == 0);
      if (aFirst != dirDesc) { s[ix] = sb; s[l] = sa; si[ix] = ib; si[l] = ia; }
    }
  }
  __syncthreads();
  for (int i = tid; i < 1024; i += 512) fA[i] = 0;
  __syncthreads();
  for (int j = tid; j < kOut; j += 512) fA[si[j]] = 1;
  __syncthreads();
  for (int i = tid; i < 1024; i += 512) si[i] = fA[i];  // keep a pristine copy
  __syncthreads();
  int* a = fA; int* b = fB;
  for (int off = 1; off < 1024; off <<= 1) {  // Hillis-Steele inclusive scan
    for (int i = tid; i < 1024; i += 512)
      b[i] = a[i] + (i >= off ? a[i - off] : 0);
    __syncthreads();
    int* t = a; a = b; b = t;
  }
  for (int i = tid; i < nPer; i += 512) {
    int gi = g * nPer + i;
    if (si[i]) {
      int np = g * kOut + a[i] - 1;
      inv[gi] = np;
      perm[np] = gi;
    } else {
      inv[gi] = -1;
    }
  }
}

// newx[nn,d] = x[perm[nn],d] * tanh(score[perm[nn]])
__global__ void k_gather(const float* __restrict__ x, const float* __restrict__ sc,
                         const int* __restrict__ perm, float* __restrict__ xo,
                         long long total) {
  long long t = blockIdx.x * (long long)blockDim.x + threadIdx.x;
  if (t >= total) return;
  int nn = (int)(t >> 6), d = (int)(t & 63);
  int o = perm[nn];
  xo[t] = x[(size_t)o * 64 + d] * tanhf(sc[o]);
}

// h[g, 0:64] (+)= max_j x[g,j,:], h[g, 64:128] (+)= mean_j x[g,j,:]
__global__ __launch_bounds__(512) void k_readout(const float* __restrict__ x,
                                                 int k, float* __restrict__ h,
                                                 int add) {
  __shared__ float smax[512];
  __shared__ float ssum[512];
  int g = blockIdx.x, tid = threadIdx.x;
  int d = tid & 63, sl = tid >> 6;  // 8 j-slices
  float mx = -3.402823466e38f, sm = 0.f;
  for (int j = sl; j < k; j += 8) {
    float v = x[((size_t)g * k + j) * 64 + d];
    mx = fmaxf(mx, v);
    sm += v;
  }
  smax[tid] = mx; ssum[tid] = sm;
  __syncthreads();
  if (sl == 0) {
    for (int s2 = 1; s2 < 8; ++s2) {
      mx = fmaxf(mx, smax[d + 64 * s2]);
      sm += ssum[d + 64 * s2];
    }
    if (add) { h[g * 128 + d] += mx; h[g * 128 + 64 + d] += sm / k; }
    else     { h[g * 128 + d]  = mx; h[g * 128 + 64 + d]  = sm / k; }
  }
}

__global__ void k_remap(const int* __restrict__ s, const int* __restrict__ d,
                        const int* __restrict__ em, const int* __restrict__ inv,
                        int* __restrict__ so, int* __restrict__ dso,
                        int* __restrict__ emo, int E) {
  int t = blockIdx.x * blockDim.x + threadIdx.x;
  if (t >= E) return;
  int a = inv[s[t]], b = inv[d[t]];
  emo[t] = em[t] & (a >= 0) & (b >= 0);
  so[t] = a < 0 ? 0 : a;
  dso[t] = b < 0 ? 0 : b;
}

__global__ void k_batch(float* __restrict__ o, int K3, int total) {
  int t = blockIdx.x * blockDim.x + threadIdx.x;
  if (t >= total) return;
  o[t] = (float)(t / K3);
}

static inline dim3 gridFor(long long t, int bs) {
  return dim3((unsigned)((t + bs - 1) / bs));
}

extern "C" void kernel_launch(void* const* d_in, const int* in_sizes, int n_in,
                              void* d_out, int out_size, void* d_ws,
                              size_t ws_size, hipStream_t stream) {
  const int B = 128, N0 = 1024, DEG = 16;
  const int N = B * N0;    // 131072
  const int E = N * DEG;   // 2097152
  const int K1 = 820, K2 = 656, K3 = 525;

  const int* x_ids = (const int*)d_in[0];
  const int* eidx = (const int*)d_in[1];
  const float* emb = (const float*)d_in[2];

  char* w = (char*)d_ws;
  auto alloc = [&](size_t bytes) {
    void* p = (void*)w;
    w += (bytes + 255) & ~(size_t)255;
    return p;
  };
  float* xA   = (float*)alloc((size_t)N * 64 * 4);
  float* xB   = (float*)alloc((size_t)N * 64 * 4);
  float* msg  = (float*)alloc((size_t)N * 64 * 4);
  float* aggr = (float*)alloc((size_t)N * 64 * 4);
  float* sc   = (float*)alloc((size_t)N * 4);
  int* inv    = (int*)alloc((size_t)N * 4);
  int* perm   = (int*)alloc((size_t)N * 4);
  int* srcA   = (int*)alloc((size_t)E * 4);
  int* dstA   = (int*)alloc((size_t)E * 4);
  int* emA    = (int*)alloc((size_t)E * 4);
  int* srcB   = (int*)alloc((size_t)E * 4);
  int* dstB   = (int*)alloc((size_t)E * 4);
  int* emB    = (int*)alloc((size_t)E * 4);
  float* h    = (float*)alloc(128 * 128 * 4);
  float* hb   = (float*)alloc(128 * 64 * 4);
  float* hc   = (float*)alloc(128 * 64 * 4);

  k_embed<<<gridFor((long long)N * 64, 256), 256, 0, stream>>>(x_ids, emb, xA,
                                                               N * 64);
  k_init_edges<<<gridFor(E, 256), 256, 0, stream>>>(eidx, eidx + E, srcA, dstA,
                                                    emA, E);

  int nIn[3] = {N, B * K1, B * K2};
  int nPer[3] = {N0, K1, K2};
  int kOut[3] = {K1, K2, K3};
  int *curS = srcA, *curD = dstA, *curM = emA;
  int *nxtS = srcB, *nxtD = dstB, *nxtM = emB;

  for (int L = 0; L < 3; ++L) {
    const float* Wl = (const float*)d_in[3 + 4 * L];
    const float* bl = (const float*)d_in[4 + 4 * L];
    const float* Ul = (const float*)d_in[5 + 4 * L];
    const float* pl = (const float*)d_in[6 + 4 * L];
    int n = nIn[L];
    dim3 gg((unsigned)((n / 16 + 7) / 8));  // 8 waves per 256-thread block
    // msg = relu(x W^T + b); dual-store also initializes aggr (self-loop max)
    k_gemm64<<<gg, 256, 0, stream>>>(xA, 64, nullptr, 0, Wl, bl, msg, aggr, n, 0);
    // aggr = max(aggr, msg[src]) over live edges (native u32 atomic max)
    k_edge_max<<<gridFor((long long)E * 16, 256), 256, 0, stream>>>(
        curS, curD, curM, msg, aggr, (long long)E * 16);
    // x2 = relu([aggr | x] @ U^T) — concat never materialized
    k_gemm64<<<gg, 256, 0, stream>>>(aggr, 64, xA, 64, Ul, nullptr, xB, nullptr,
                                     n, 0);
    k_score<<<gridFor((long long)n * 32, 256), 256, 0, stream>>>(xB, pl, sc, n);
    k_topk<<<dim3(B), 512, 0, stream>>>(sc, nPer[L], kOut[L], inv, perm);
    int newN = B * kOut[L];
    k_gather<<<gridFor((long long)newN * 64, 256), 256, 0, stream>>>(
        xB, sc, perm, xA, (long long)newN * 64);
    k_readout<<<dim3(B), 512, 0, stream>>>(xA, kOut[L], h, L > 0 ? 1 : 0);
    if (L < 2) {
      k_remap<<<gridFor(E, 256), 256, 0, stream>>>(curS, curD, curM, inv, nxtS,
                                                   nxtD, nxtM, E);
      int* t;
      t = curS; curS = nxtS; nxtS = t;
      t = curD; curD = nxtD; nxtD = t;
      t = curM; curM = nxtM; nxtM = t;
    }
  }

  // MLP head: 128 rows = 8 row-tiles = one 256-thread block each
  k_gemm64<<<dim3(1), 256, 0, stream>>>(h, 128, nullptr, 0,
                                        (const float*)d_in[15],
                                        (const float*)d_in[16], hb, nullptr,
                                        128, 0);
  k_gemm64<<<dim3(1), 256, 0, stream>>>(hb, 64, nullptr, 0,
                                        (const float*)d_in[17],
                                        (const float*)d_in[18], hc, nullptr,
                                        128, 0);
  k_gemm64<<<dim3(1), 256, 0, stream>>>(hc, 64, nullptr, 0,
                                        (const float*)d_in[19],
                                        (const float*)d_in[20], (float*)d_out,
                                        nullptr, 128, 1);  // sigmoid
  k_batch<<<gridFor((long long)B * K3, 256), 256, 0, stream>>>(
      (float*)d_out + B * 64, K3, B * K3);
}